// ExpertLinear_10737418240132
// MI455X (gfx1250) — compile-verified
//
#include <hip/hip_runtime.h>

typedef __attribute__((ext_vector_type(2))) float v2f;
typedef __attribute__((ext_vector_type(8))) float v8f;
typedef __attribute__((__vector_size__(4 * sizeof(int)))) int v4i;

#define B_TOT   4096
#define K_EXP   8
#define IN_DIM  1024
#define OUT_DIM 1024

#define BM 128           // batch-rows per block tile
#define BN 128           // out-cols per block tile
#define KI 16            // i-dim chunk per pipeline stage
#define LSTRIDE 20       // 16 floats + 4 pad: bank-conflict-free, 16B-aligned rows
#define NSTAGE (IN_DIM / KI)   // 64 stages
#define LOADS_PER_STAGE 18     // per-thread async b128 copies per stage (2 X + 16 W)

// ---- gfx1250 async global->LDS path (ASYNCcnt), with portable fallback ----
#if defined(__has_builtin)
#if __has_builtin(__builtin_amdgcn_global_load_async_to_lds_b128) && \
    __has_builtin(__builtin_amdgcn_s_wait_asynccnt)
#define USE_ASYNC_LDS 1
#endif
#endif
#ifndef USE_ASYNC_LDS
#define USE_ASYNC_LDS 0
#endif

typedef v4i __attribute__((address_space(1)))* g_v4i_ptr;   // global (AS1)
typedef v4i __attribute__((address_space(3)))* l_v4i_ptr;   // LDS (AS3)

__device__ __forceinline__ void copy16_g2l(const float* __restrict__ g, float* l) {
#if USE_ASYNC_LDS
    // Per-lane async copy of 16B, global -> LDS, no VGPR round-trip.
    // Low 32 bits of a generic LDS address are the LDS byte offset on CDNA5.
    __builtin_amdgcn_global_load_async_to_lds_b128(
        (g_v4i_ptr)(unsigned long long)g,
        (l_v4i_ptr)(unsigned int)(unsigned long long)l,
        0, 0);
#else
    *(float4*)l = *(const float4*)g;
#endif
}

__device__ __forceinline__ void wait_async_keep_next_stage() {
#if USE_ASYNC_LDS
    // Two stages in flight (2*18): allow the newest stage's 18 to remain
    // outstanding; in-order completion => previous stage has fully landed.
    __builtin_amdgcn_s_wait_asynccnt(LOADS_PER_STAGE);
#endif
}

__device__ __forceinline__ void wait_async_all() {
#if USE_ASYNC_LDS
    __builtin_amdgcn_s_wait_asynccnt(0);
#endif
}

__global__ __launch_bounds__(256) void moe_blend_gemm(
    const float* __restrict__ x,        // (B, IN)
    const float* __restrict__ blendw,   // (B, K)
    const float* __restrict__ weight,   // (K, OUT, IN)
    const float* __restrict__ bias,     // (K, OUT)
    float* __restrict__ out)            // (B, OUT)
{
    __shared__ float Xs[2][BM * LSTRIDE];            // 2 x 10 KB
    __shared__ float Ws[2][K_EXP][BN * LSTRIDE];     // 2 x 80 KB
    __shared__ float blend_s[BM * K_EXP];            // 4 KB
    __shared__ float bias_s[K_EXP * BN];             // 4 KB

    const int t     = threadIdx.x;
    const int lane  = t & 31;          // wave32
    const int wave  = t >> 5;          // 8 waves
    const int wrow  = wave >> 2;       // 0..1  -> 64-row band
    const int wcol  = wave & 3;        // 0..3  -> 32-col band
    const int nlane = lane & 15;
    const int half  = lane >> 4;
    const int bBase = blockIdx.y * BM;
    const int oBase = blockIdx.x * BN;

    // Stage blend tile (128x8) and bias tile (8x128) once per block.
    {
        int r = t >> 1, p = t & 1;
        *(float4*)&blend_s[r * K_EXP + p * 4] =
            *(const float4*)&blendw[(size_t)(bBase + r) * K_EXP + p * 4];
        int k = t >> 5, c4 = t & 31;
        *(float4*)&bias_s[k * BN + c4 * 4] =
            *(const float4*)&bias[(size_t)k * OUT_DIM + oBase + c4 * 4];
    }

    // Issue one pipeline stage of async copies: X tile (128 x 16) + 8 W tiles.
    auto stage_loads = [&](int s, int buf) {
        const int i0 = s * KI;
        for (int j = 0; j < 2; ++j) {                       // 2 async b128
            int idx = t + 256 * j;
            int row = idx >> 2, c4 = idx & 3;
            copy16_g2l(&x[(size_t)(bBase + row) * IN_DIM + i0 + c4 * 4],
                       &Xs[buf][row * LSTRIDE + c4 * 4]);
        }
        for (int k = 0; k < K_EXP; ++k) {                   // 16 async b128
            for (int j = 0; j < 2; ++j) {
                int idx = t + 256 * j;
                int row = idx >> 2, c4 = idx & 3;
                copy16_g2l(&weight[((size_t)k * OUT_DIM + oBase + row) * IN_DIM
                                   + i0 + c4 * 4],
                           &Ws[buf][k][row * LSTRIDE + c4 * 4]);
            }
        }
    };

    v8f acc[4][2];
    const v8f vzero = {0.f, 0.f, 0.f, 0.f, 0.f, 0.f, 0.f, 0.f};
    for (int fr = 0; fr < 4; ++fr)
        for (int fc = 0; fc < 2; ++fc)
            acc[fr][fc] = vzero;

    stage_loads(0, 0);                 // prime the pipeline

    for (int s = 0; s < NSTAGE; ++s) {
        const int buf = s & 1;
        if (s + 1 < NSTAGE) {
            stage_loads(s + 1, buf ^ 1);   // prefetch next stage (async)
            wait_async_keep_next_stage();  // stage s has landed in LDS
        } else {
            wait_async_all();
        }
        __syncthreads();

        // ---- compute: all 8 experts of this stage into one accumulator ----
        for (int k = 0; k < K_EXP; ++k) {
            // Per-lane blend scalar: A-frag lane holds row M = lane%16.
            float bl[4];
            for (int fr = 0; fr < 4; ++fr)
                bl[fr] = blend_s[(wrow * 64 + fr * 16 + nlane) * K_EXP + k];

            for (int kk = 0; kk < KI; kk += 4) {
                // f32 16x16x4 A layout: lanes 0-15 K={0,1}, lanes 16-31 K={2,3}
                const int c0 = kk + 2 * half;
                v2f a[4], b[2];
                for (int fr = 0; fr < 4; ++fr) {
                    int row = wrow * 64 + fr * 16 + nlane;
                    v2f av = *(const v2f*)&Xs[buf][row * LSTRIDE + c0];
                    a[fr] = av * bl[fr];           // fold blend into A rows
                }
                for (int fc = 0; fc < 2; ++fc) {
                    int col = wcol * 32 + fc * 16 + nlane;
                    b[fc] = *(const v2f*)&Ws[buf][k][col * LSTRIDE + c0];
                }
                for (int fr = 0; fr < 4; ++fr)
                    for (int fc = 0; fc < 2; ++fc)
                        acc[fr][fc] = __builtin_amdgcn_wmma_f32_16x16x4_f32(
                            false, a[fr], false, b[fc],
                            (short)0, acc[fr][fc], false, false);
            }
        }
        __syncthreads();   // protect buf before stage s+2 overwrites it
    }

    // ---- epilogue: add blend@bias term and store ----
    // C layout: vgpr r, lane l -> M = r + 8*(l>=16), N = l%16
    for (int fr = 0; fr < 4; ++fr) {
        for (int fc = 0; fc < 2; ++fc) {
            const int colL = wcol * 32 + fc * 16 + nlane;
            const int o    = oBase + colL;
            for (int r = 0; r < 8; ++r) {
                const int rowL = wrow * 64 + fr * 16 + r + 8 * half;
                float v = acc[fr][fc][r];
                for (int k = 0; k < K_EXP; ++k)
                    v += blend_s[rowL * K_EXP + k] * bias_s[k * BN + colL];
                out[(size_t)(bBase + rowL) * OUT_DIM + o] = v;
            }
        }
    }
}

extern "C" void kernel_launch(void* const* d_in, const int* in_sizes, int n_in,
                              void* d_out, int out_size, void* d_ws, size_t ws_size,
                              hipStream_t stream) {
    (void)in_sizes; (void)n_in; (void)out_size; (void)d_ws; (void)ws_size;
    const float* x    = (const float*)d_in[0];
    const float* bw   = (const float*)d_in[1];
    const float* w    = (const float*)d_in[2];
    const float* bias = (const float*)d_in[3];
    float* out        = (float*)d_out;

    dim3 grid(OUT_DIM / BN, B_TOT / BM);   // (8, 32) = 256 workgroups
    moe_blend_gemm<<<grid, dim3(256), 0, stream>>>(x, bw, w, bias, out);
}